// TTN1_BP_40638980555123
// MI455X (gfx1250) — compile-verified
//
#include <hip/hip_runtime.h>

typedef __attribute__((ext_vector_type(2))) float v2f;
typedef __attribute__((ext_vector_type(8))) float v8f;

#define NUMN  64     // batch
#define CIN   4      // input channels
#define SY    256    // y extent of x
#define PLANE 65536  // SX*SY plane of one (n,c,dd)
#define NXX   128    // output x positions
#define NYY   128    // output y positions
#define TY    8      // y-positions (waves) per workgroup

// One wave per output position (X,Y): M=64(n) x K=64(c*16+q) x N=8(p*2+o, padded to 16)
// GEMM via V_WMMA_F32_16X16X4_F32: 4 M-tiles x 16 K-steps = 64 WMMAs per wave.
__global__ __launch_bounds__(256, 1)
void ttn1_wmma_kernel(const float* __restrict__ x,
                      const float* __restrict__ w,
                      const float* __restrict__ bias,
                      float* __restrict__ out) {
    __shared__ float s_out[NUMN * 8 * TY];   // [n][p*2+o][y]
    __shared__ float s_bias[4 * TY * 2];     // [p][y][o]

    const int tid  = threadIdx.x;
    const int lane = tid & 31;
    const int wv   = tid >> 5;        // wave id == local y position
    const int X    = blockIdx.x;
    const int Y0   = blockIdx.y * TY;
    const int Y    = Y0 + wv;
    const int row  = lane & 15;       // M row within tile / N column
    const int hk   = lane >> 4;       // half-wave: selects K-slot pair (k index of e)

    // Stage bias slice for this (X, Y0..Y0+7) once per workgroup.
    if (tid < 64) {
        const int p = tid >> 4, yy = (tid >> 1) & 7, o = tid & 1;
        s_bias[tid] = bias[((p * NXX + X) * NYY + (Y0 + yy)) * 2 + o];
    }

    // ---- B-fragment preload: lane column N=row -> (p,o); N>=8 pads with zero ----
    const bool nvalid = (row < 8);
    const int  p = row >> 1, o = row & 1;
    float B0[16], B1[16];
#pragma unroll
    for (int t = 0; t < 16; ++t) {
        const int c  = t >> 2;
        const int q0 = (t & 3) * 4 + hk * 2;   // K-slot base for this half-wave
        if (nvalid) {
            const float* wb = w + ((((c * 4 + p) * NXX + X) * NYY + Y) * 16 + q0) * 2 + o;
            B0[t] = wb[0];     // q0,   (v0: K=4t + 2*hk)
            B1[t] = wb[2];     // q0+1, (v1: K=4t + 2*hk + 1)
        } else {
            B0[t] = 0.0f; B1[t] = 0.0f;
        }
    }

    v8f acc[4] = {};   // 4 M-tiles of 16x16 f32 accumulators

    const int xy = 2 * X * SY + 2 * Y;   // (2X, 2Y) within one plane

#pragma unroll
    for (int c = 0; c < CIN; ++c) {
        // Corner loads for this lane's 4 n values (M-tiles):
        //  L0=(a0,e0) @ (dd=0,2X)  L1=(b0,f0) @ (dd=0,2X+1)
        //  L2=(a1,e1) @ (dd=1,2X)  L3=(b1,f1) @ (dd=1,2X+1)
        float2 L0[4], L1[4], L2[4], L3[4];
#pragma unroll
        for (int m = 0; m < 4; ++m) {
            const int n = m * 16 + row;
            const float* bp = x + (size_t)((n * CIN + c) * 2) * PLANE + xy;
            L0[m] = *(const float2*)(bp);
            L1[m] = *(const float2*)(bp + SY);
            L2[m] = *(const float2*)(bp + PLANE);
            L3[m] = *(const float2*)(bp + PLANE + SY);
        }
#pragma unroll
        for (int tt = 0; tt < 4; ++tt) {   // K-step t = c*4+tt; q = i8+j4+k2+l, (i,j)=tt bits
            const int t = c * 4 + tt;
            v2f bfrag;
            bfrag.x = B0[t];
            bfrag.y = B1[t];
#pragma unroll
            for (int m = 0; m < 4; ++m) {
                const float av = (tt & 2) ? L2[m].x : L0[m].x;  // a_i, i = tt>>1
                const float bv = (tt & 1) ? L3[m].x : L1[m].x;  // b_j, j = tt&1
                const float ev = hk ? L2[m].y : L0[m].y;        // e_k, k = half-wave
                const float g  = av * bv * ev;
                v2f afrag;
                afrag.x = g * L1[m].y;   // * f0  (l=0, K-slot v0)
                afrag.y = g * L3[m].y;   // * f1  (l=1, K-slot v1)
                acc[m] = __builtin_amdgcn_wmma_f32_16x16x4_f32(
                    false, afrag, false, bfrag, (short)0, acc[m], false, false);
            }
        }
    }

    // D layout: lane<16: VGPR r -> (n = m*16 + r,     N = row)
    //           lane>=16: VGPR r -> (n = m*16 + 8 + r, N = row)
    if (nvalid) {
#pragma unroll
        for (int m = 0; m < 4; ++m) {
            const int nb = m * 16 + hk * 8;
#pragma unroll
            for (int r = 0; r < 8; ++r) {
                s_out[((nb + r) * 8 + row) * TY + wv] = acc[m][r];
            }
        }
    }
    __syncthreads();

    // Coalesced output: each combo (n, N=p*2+o) owns 8 consecutive Y floats (32B).
    for (int cb = tid; cb < NUMN * 8; cb += 256) {
        const int N  = cb & 7;
        const int pp = N >> 1, oo = N & 1;
        float4 va, vb;
        va.x = s_out[cb * 8 + 0] + s_bias[pp * 16 + 0  + oo];
        va.y = s_out[cb * 8 + 1] + s_bias[pp * 16 + 2  + oo];
        va.z = s_out[cb * 8 + 2] + s_bias[pp * 16 + 4  + oo];
        va.w = s_out[cb * 8 + 3] + s_bias[pp * 16 + 6  + oo];
        vb.x = s_out[cb * 8 + 4] + s_bias[pp * 16 + 8  + oo];
        vb.y = s_out[cb * 8 + 5] + s_bias[pp * 16 + 10 + oo];
        vb.z = s_out[cb * 8 + 6] + s_bias[pp * 16 + 12 + oo];
        vb.w = s_out[cb * 8 + 7] + s_bias[pp * 16 + 14 + oo];
        float* op = out + (size_t)cb * (NXX * NYY) + X * NYY + Y0;
        *(float4*)(op)     = va;
        *(float4*)(op + 4) = vb;
    }
}

extern "C" void kernel_launch(void* const* d_in, const int* in_sizes, int n_in,
                              void* d_out, int out_size, void* d_ws, size_t ws_size,
                              hipStream_t stream) {
    (void)in_sizes; (void)n_in; (void)out_size; (void)d_ws; (void)ws_size;
    const float* x  = (const float*)d_in[0];
    const float* w  = (const float*)d_in[1];
    const float* b  = (const float*)d_in[2];
    float* out      = (float*)d_out;
    dim3 grid(NXX, NYY / TY, 1);
    ttn1_wmma_kernel<<<grid, dim3(256, 1, 1), 0, stream>>>(x, w, b, out);
}